// Blur_82471962018173
// MI455X (gfx1250) — compile-verified
//
#include <hip/hip_runtime.h>
#include <hip/hip_bf16.h>

// Blur (stylegan2 upfirdn2d, up=1,down=1,pad=(1,1)) on (4,128,513,513)f32 -> (4,128,512,512)f32.
// Memory-bound: ~1.08 GB traffic -> ~46us floor @ 23.3 TB/s. Separable rank-1 FIR
// (af[i]*bf[j] == flipped kernel). Vertical pass uses packed f32 (v_pk_fma_f32),
// horizontal pass scalar FMA (VOPD dual-issue pairs them).
// CDNA5 path: GLOBAL_LOAD_ASYNC_TO_LDS_B32 streams the input tile into LDS
// (ASYNCcnt, s_wait_asynccnt); coefficient math overlaps the DMA.

#define W_IN   513
#define W_OUT  512
#define TILE    64          // output tile (64x64) per workgroup
#define IH      67          // input tile rows/cols needed (64 + 3 halo)
#define LSTRIDE 68          // LDS row stride in floats (rows 16B aligned)
#define NLOAD  (IH * LSTRIDE)
#define BLOCK  256

typedef float v4f __attribute__((ext_vector_type(4)));
typedef float v2f __attribute__((ext_vector_type(2)));

__global__ __launch_bounds__(BLOCK) void blur_sep_kernel(
    const float* __restrict__ x,     // (NC, 513, 513)
    const float* __restrict__ kern,  // (4,4) FIR buffer
    float* __restrict__ out)         // (NC, 512, 512)
{
    __shared__ __align__(16) float tile[NLOAD];

    const int tid = threadIdx.x;
    const int wgx = blockIdx.x, wgy = blockIdx.y, img = blockIdx.z;

    const float* imgIn  = x   + (size_t)img * (size_t)(W_IN  * W_IN);
    float*       imgOut = out + (size_t)img * (size_t)(W_OUT * W_OUT);

    const int gx0 = wgx * TILE - 1;   // top-left of input tile (may be -1)
    const int gy0 = wgy * TILE - 1;

    const unsigned ldsBase = (unsigned)(uintptr_t)tile;  // LDS byte offset (addr[31:0])

    // ---- Issue async DMA of input tile into LDS; zero OOB halo via ds_store ----
    // Incremental (lr,lc) bookkeeping: step of 256 = +3 rows + 52 cols (one wrap max).
    {
        int lr = tid / LSTRIDE;
        int lc = tid - lr * LSTRIDE;
        for (int i = tid; i < NLOAD; i += BLOCK) {
            const int gr = gy0 + lr;
            const int gc = gx0 + lc;
            if (gr >= 0 && gr < W_IN && gc >= 0 && gc < W_IN && lc < IH) {
                const float* gp = imgIn + gr * W_IN + gc;
                const unsigned la = ldsBase + (unsigned)i * 4u;
                // per-lane async DMA: global -> LDS, tracked by ASYNCcnt
                asm volatile("global_load_async_to_lds_b32 %0, %1, off"
                             :
                             : "v"(la), "v"(gp)
                             : "memory");
            } else {
                tile[i] = 0.0f;   // zero-pad halo (and unused col 67)
            }
            lc += 256 - 3 * LSTRIDE;   // +52
            lr += 3;
            if (lc >= LSTRIDE) { lc -= LSTRIDE; lr += 1; }
        }
    }

    // ---- Rank-1 factorization of the flipped 4x4 kernel (overlaps the DMA) ----
    // kflip[i][j] = K[3-i][3-j]; af[i]*bf[j] == kflip[i][j] exactly for rank-1 K.
    const float is = 1.0f / sqrtf(kern[15]);   // 1/sqrt(K[3][3])
    float af[4], bf[4];
#pragma unroll
    for (int i = 0; i < 4; ++i) af[i] = kern[15 - 4 * i] * is;  // K[3-i][3] / s
#pragma unroll
    for (int j = 0; j < 4; ++j) bf[j] = kern[15 - j] * is;      // K[3][3-j] / s

    asm volatile("s_wait_asynccnt 0x0" ::: "memory");  // this wave's async DMAs done
    __syncthreads();                                    // all waves' LDS writes visible

    // ---- Each thread: 4x4 output patch ----
    const int tx = tid & 15;     // 16 threads across
    const int ty = tid >> 4;     // 16 threads down
    const int c0 = tx * 4;       // tile-local output col
    const int r0 = ty * 4;       // tile-local output row

    // Load 7 rows x 8 cols of input tile (need 7x7, 8th col unused) as b128 pairs,
    // kept as column-pair vectors for packed math.
    v2f rr2[7][4];
#pragma unroll
    for (int i = 0; i < 7; ++i) {
        const v4f a0 = *(const v4f*)&tile[(r0 + i) * LSTRIDE + c0];
        const v4f a1 = *(const v4f*)&tile[(r0 + i) * LSTRIDE + c0 + 4];
        rr2[i][0] = __builtin_shufflevector(a0, a0, 0, 1);
        rr2[i][1] = __builtin_shufflevector(a0, a0, 2, 3);
        rr2[i][2] = __builtin_shufflevector(a1, a1, 0, 1);
        rr2[i][3] = __builtin_shufflevector(a1, a1, 2, 3);
    }

    // Vertical 4-tap pass, packed 2xf32 (v_pk_fma_f32): t2[r][p] = sum_i af[i]*rr2[r+i][p]
    v2f t2[4][4];
#pragma unroll
    for (int r = 0; r < 4; ++r) {
#pragma unroll
        for (int p = 0; p < 4; ++p) {
            v2f v = (v2f)(af[0]) * rr2[r][p];
            v = __builtin_elementwise_fma((v2f)(af[1]), rr2[r + 1][p], v);
            v = __builtin_elementwise_fma((v2f)(af[2]), rr2[r + 2][p], v);
            v = __builtin_elementwise_fma((v2f)(af[3]), rr2[r + 3][p], v);
            t2[r][p] = v;
        }
    }

    // Horizontal 4-tap pass (scalar FMA, shifted windows) + non-temporal b128 stores.
    const int oxg = wgx * TILE + c0;
    const int oyg = wgy * TILE + r0;
#pragma unroll
    for (int r = 0; r < 4; ++r) {
        v4f o;
#pragma unroll
        for (int c = 0; c < 4; ++c) {
            float v = bf[0] * t2[r][(c) >> 1][(c) & 1];
            v = fmaf(bf[1], t2[r][(c + 1) >> 1][(c + 1) & 1], v);
            v = fmaf(bf[2], t2[r][(c + 2) >> 1][(c + 2) & 1], v);
            v = fmaf(bf[3], t2[r][(c + 3) >> 1][(c + 3) & 1], v);
            o[c] = v;
        }
        __builtin_nontemporal_store(o, (v4f*)(imgOut + (size_t)(oyg + r) * W_OUT + oxg));
    }
}

extern "C" void kernel_launch(void* const* d_in, const int* in_sizes, int n_in,
                              void* d_out, int out_size, void* d_ws, size_t ws_size,
                              hipStream_t stream) {
    const float* x = (const float*)d_in[0];
    const float* k = (const float*)d_in[1];
    float* out = (float*)d_out;
    (void)d_ws; (void)ws_size; (void)n_in; (void)out_size;

    const int nc = in_sizes[0] / (W_IN * W_IN);   // 4*128 = 512 images
    dim3 grid(W_OUT / TILE, W_OUT / TILE, nc);    // (8, 8, 512)
    blur_sep_kernel<<<grid, dim3(BLOCK), 0, stream>>>(x, k, out);
}